// Homography_19713899889337
// MI455X (gfx1250) — compile-verified
//
#include <hip/hip_runtime.h>

typedef __attribute__((ext_vector_type(2))) float v2f;
typedef __attribute__((ext_vector_type(8))) float v8f;

#define IMG_B 32
#define IMG_H 512
#define IMG_W 512
#define IMG_C 3
#define ROW_ELTS (IMG_W * IMG_C)    /* 1536 floats per image row   */
#define IMG_ELTS (IMG_H * ROW_ELTS) /* 786432 floats per batch img */

// One wave32 computes one 16x16 output tile.
//   3x V_WMMA_F32_16X16X4_F32 produce the tile's (tx, ty, tw) homography maps:
//   D[r,c] = base + H[i][0]*c + H[i][1]*r   via   A[r,:]=(1, r, 0, 0),
//   B[:,c]=(base + H[i][0]*c, H[i][1], 0, 0).
//   D layout: lane L holds column N = L&15; VGPR v holds row M = v + 8*(L>=16),
//   so each lane owns 8 pixels of a column -> full-wave gather phase.
// Divergent addressing: ONE unsigned 32-bit offset per corner folded as
// saddr + voffset(scale_offset); channels are constant-ioffset loads off that
// pointer so they merge into a single global_load_b96. Output stores are
// non-temporal so the 100MB input stays resident in the 192MB L2.
__global__ __launch_bounds__(256) void homography_warp_kernel(
    const float* __restrict__ images,
    const float* __restrict__ homos,
    float* __restrict__ out)
{
    const int lane = threadIdx.x & 31;
    const int wave = threadIdx.x >> 5;
    const int b = blockIdx.z;                       // wave-uniform -> s_load of H
    const int tile_x0 = blockIdx.x * 16;
    const int tile_y0 = (blockIdx.y * 8 + wave) * 16;

    const float* Hm = homos + b * 9;
    const float h00 = Hm[0], h01 = Hm[1], h02 = Hm[2];
    const float h10 = Hm[3], h11 = Hm[4], h12 = Hm[5];
    const float h20 = Hm[6], h21 = Hm[7], h22 = Hm[8];

    const float fx0 = (float)tile_x0, fy0 = (float)tile_y0;
    const float base_x = h00 * fx0 + h01 * fy0 + h02;
    const float base_y = h10 * fx0 + h11 * fy0 + h12;
    const float base_w = h20 * fx0 + h21 * fy0 + h22;

    const bool lo = lane < 16;          // lanes 0-15 carry K=0,1; K=2,3 zeroed
    const float n = (float)(lane & 15); // column index for B

    v2f a;                              // A 16x4: A[r,0]=1, A[r,1]=r
    a.x = lo ? 1.0f : 0.0f;
    a.y = lo ? (float)lane : 0.0f;

    v2f bx, by, bw;                     // B 4x16 per coordinate map
    bx.x = lo ? (base_x + h00 * n) : 0.0f;  bx.y = lo ? h01 : 0.0f;
    by.x = lo ? (base_y + h10 * n) : 0.0f;  by.y = lo ? h11 : 0.0f;
    bw.x = lo ? (base_w + h20 * n) : 0.0f;  bw.y = lo ? h21 : 0.0f;

    const v8f zero = {};
    // (neg_a, A, neg_b, B, c_mod, C, reuse_a, reuse_b)
    v8f tx = __builtin_amdgcn_wmma_f32_16x16x4_f32(false, a, false, bx,
                                                   (short)0, zero, false, false);
    v8f ty = __builtin_amdgcn_wmma_f32_16x16x4_f32(false, a, false, by,
                                                   (short)0, zero, false, false);
    v8f tw = __builtin_amdgcn_wmma_f32_16x16x4_f32(false, a, false, bw,
                                                   (short)0, zero, false, false);

    const int rbase = (lane >> 4) * 8;      // row offset within tile
    const int c = lane & 15;                // column within tile
    const unsigned col_off = (unsigned)(tile_x0 + c) * IMG_C;

    // Wave-uniform 64-bit bases; per-lane offsets are unsigned 32-bit.
    const float* img  = images + (size_t)b * IMG_ELTS;
    float*       outb = out    + (size_t)b * IMG_ELTS;

#pragma unroll
    for (int v = 0; v < 8; ++v) {
        const int r = rbase + v;
        // perspective divide: hw rcp + 1 Newton-Raphson step (~1 ulp)
        const float w  = tw[v];
        const float r0 = __builtin_amdgcn_rcpf(w);
        const float invw = fmaf(fmaf(-w, r0, 1.0f), r0, r0);
        const float x = tx[v] * invw;
        const float y = ty[v] * invw;

        const float x0f = floorf(x);
        const float y0f = floorf(y);

        // border clamp -> [0, 511]; unsigned so zext folds into saddr voffset
        const unsigned x0 = (unsigned)(int)fminf(fmaxf(x0f,        0.0f), (float)(IMG_W - 1));
        const unsigned x1 = (unsigned)(int)fminf(fmaxf(x0f + 1.0f, 0.0f), (float)(IMG_W - 1));
        const unsigned y0 = (unsigned)(int)fminf(fmaxf(y0f,        0.0f), (float)(IMG_H - 1));
        const unsigned y1 = (unsigned)(int)fminf(fmaxf(y0f + 1.0f, 0.0f), (float)(IMG_H - 1));

        const unsigned row0 = y0 * ROW_ELTS;
        const unsigned row1 = y1 * ROW_ELTS;
        // ONE zext'd offset per corner; channels are constant ioffsets -> b96
        const float* pa = img + (row0 + x0 * IMG_C);   // (y0,x0)
        const float* pb = img + (row1 + x0 * IMG_C);   // (y1,x0)
        const float* pc = img + (row0 + x1 * IMG_C);   // (y0,x1)
        const float* pd = img + (row1 + x1 * IMG_C);   // (y1,x1)

        // issue all 4 corner gathers before any arithmetic uses them
        const float a0 = pa[0], a1 = pa[1], a2 = pa[2];
        const float b0 = pb[0], b1 = pb[1], b2 = pb[2];
        const float c0 = pc[0], c1 = pc[1], c2 = pc[2];
        const float d0 = pd[0], d1 = pd[1], d2 = pd[2];

        const float fx = x - x0f;
        const float fy = y - y0f;
        const float wa = (1.0f - fx) * (1.0f - fy);
        const float wb = (1.0f - fx) * fy;
        const float wc = fx * (1.0f - fy);
        const float wd = fx * fy;

        float* po = outb + ((unsigned)(tile_y0 + r) * ROW_ELTS + col_off);
        __builtin_nontemporal_store(wa * a0 + wb * b0 + wc * c0 + wd * d0, po + 0);
        __builtin_nontemporal_store(wa * a1 + wb * b1 + wc * c1 + wd * d1, po + 1);
        __builtin_nontemporal_store(wa * a2 + wb * b2 + wc * c2 + wd * d2, po + 2);
    }
}

extern "C" void kernel_launch(void* const* d_in, const int* in_sizes, int n_in,
                              void* d_out, int out_size, void* d_ws, size_t ws_size,
                              hipStream_t stream) {
    const float* images = (const float*)d_in[0];   // [32,512,512,3] fp32
    const float* homos  = (const float*)d_in[1];   // [32,3,3] fp32
    float* out = (float*)d_out;                    // [32,512,512,3] fp32

    dim3 block(256);                                // 8 waves of 32
    dim3 grid(IMG_W / 16, IMG_H / (16 * 8), IMG_B); // (32, 4, 32)
    homography_warp_kernel<<<grid, block, 0, stream>>>(images, homos, out);
}